// HyperbolicResidualAdd_47777216201295
// MI455X (gfx1250) — compile-verified
//
#include <hip/hip_runtime.h>
#include <math.h>
#include <stdint.h>

// Flip to 0 if the gfx1250 assembler rejects the async-to-LDS mnemonic.
#define USE_ASYNC_LDS 1

namespace {

constexpr int D      = 768;               // feature dim (64 x 577 rows)
constexpr int WPB    = 8;                 // waves (= rows) per block
constexpr int LANES  = 32;                // wave32
constexpr int CHUNKS = D / (LANES * 4);   // 6 float4 chunks per lane

// ---- coefficient algebra over span{x, y} -------------------------------
struct V2   { float a, b; };
struct Gram { float X2, Y2, XY; };

__device__ inline float dot2(const V2& v, const V2& w, const Gram& g) {
  return v.a * w.a * g.X2 + v.b * w.b * g.Y2 + (v.a * w.b + w.a * v.b) * g.XY;
}
__device__ inline float nrm2(const V2& v, const Gram& g) {
  return v.a * v.a * g.X2 + v.b * v.b * g.Y2 + 2.0f * v.a * v.b * g.XY;
}
__device__ inline float cnorm(const V2& v, const Gram& g) {
  return fmaxf(sqrtf(fmaxf(nrm2(v, g), 0.0f)), 1e-15f);   // _norm with EPS clamp
}
__device__ inline V2 scl(const V2& v, float k) { return V2{v.a * k, v.b * k}; }

__device__ inline V2 clipn(const V2& v, const Gram& g, float maxn) {
  float n = cnorm(v, g);
  return scl(v, fminf(1.0f, maxn / n));
}
__device__ inline float atanh_c(float z) {        // arctanh(min(z, ATANH_MAX))
  return atanhf(fminf(z, 1.0f - 1e-7f));
}
__device__ inline V2 expmap0_c(const V2& v, const Gram& g, float sc) {
  float a = sc * cnorm(v, g);
  return scl(v, tanhf(a) / a);
}
__device__ inline V2 logmap0_c(const V2& v, const Gram& g, float sc) {
  float a = sc * cnorm(v, g);
  return scl(v, atanh_c(a) / a);
}
__device__ inline V2 msm(float r, const V2& v, const Gram& g, float sc) {
  float a = sc * cnorm(v, g);
  return scl(v, tanhf(r * atanh_c(a)) / a);
}
__device__ inline V2 mobius_add_c(const V2& v, const V2& w, const Gram& g, float c) {
  float x2 = nrm2(v, g);
  float y2 = nrm2(w, g);
  float xy = dot2(v, w, g);
  float t  = 1.0f + 2.0f * c * xy;
  float P  = t + c * y2;
  float Q  = 1.0f - c * x2;
  float inv = 1.0f / fmaxf(t + c * c * x2 * y2, 1e-15f);
  return V2{(P * v.a + Q * w.a) * inv, (P * v.b + Q * w.b) * inv};
}

// ---- wave32 butterfly reduction via ds_swizzle -------------------------
template <int M>
__device__ inline float xor_add(float v) {
  int s = __builtin_amdgcn_ds_swizzle(__float_as_int(v), 0x7C00 | M); // and=0x1f, xor=M
  return v + __int_as_float(s);
}
__device__ inline float wave_allsum(float v) {
  v = xor_add<16>(v); v = xor_add<8>(v); v = xor_add<4>(v);
  v = xor_add<2>(v);  v = xor_add<1>(v);
  return v;
}

__global__ __launch_bounds__(WPB * LANES)
void hra_kernel(const float* __restrict__ x, const float* __restrict__ y,
                const float* __restrict__ curv_hra, const float* __restrict__ gamma_raw,
                const float* __restrict__ gamma_scale, const float* __restrict__ scale_center,
                float* __restrict__ out, int rows) {
  __shared__ __align__(16) float sh[2 * WPB * D];  // [x rows | y rows]
  __shared__ float gram_sh[WPB][3];
  __shared__ float coef_sh[WPB][2];

  const int tid  = (int)threadIdx.x;
  const int wave = tid >> 5;
  const int lane = tid & 31;
  const int row  = (int)blockIdx.x * WPB + wave;
  const bool live = (row < rows);

  const int xoff = wave * D;
  const int yoff = WPB * D + wave * D;
  const float* xr = x + (size_t)row * D;
  const float* yr = y + (size_t)row * D;

  // ---------- stage both rows into LDS ----------------------------------
  if (live) {
#if USE_ASYNC_LDS
    const uint32_t lds_base = (uint32_t)(uintptr_t)(&sh[0]);
#pragma unroll
    for (int j = 0; j < CHUNKS; ++j) {
      const int e = j * 128 + lane * 4;
      uint32_t lx = lds_base + 4u * (uint32_t)(xoff + e);
      uint32_t ly = lds_base + 4u * (uint32_t)(yoff + e);
      unsigned long long gx = (unsigned long long)(uintptr_t)(xr + e);
      unsigned long long gy = (unsigned long long)(uintptr_t)(yr + e);
      asm volatile("global_load_async_to_lds_b128 %0, %1, off"
                   :: "v"(lx), "v"(gx) : "memory");
      asm volatile("global_load_async_to_lds_b128 %0, %1, off"
                   :: "v"(ly), "v"(gy) : "memory");
    }
    asm volatile("s_wait_asynccnt 0" ::: "memory");
#else
#pragma unroll
    for (int j = 0; j < CHUNKS; ++j) {
      const int e = j * 128 + lane * 4;
      float4 vx = *(const float4*)(xr + e);
      float4 vy = *(const float4*)(yr + e);
      *(float4*)&sh[xoff + e] = vx;
      *(float4*)&sh[yoff + e] = vy;
    }
#endif
  }

  // ---------- per-row Gram: X2, Y2, XY (wave-local, no barrier needed) ---
  if (live) {
    float sx2 = 0.0f, sy2 = 0.0f, sxy = 0.0f;
#pragma unroll
    for (int j = 0; j < CHUNKS; ++j) {
      const int e = j * 128 + lane * 4;
      float4 vx = *(const float4*)&sh[xoff + e];
      float4 vy = *(const float4*)&sh[yoff + e];
      sx2 = fmaf(vx.x, vx.x, sx2); sx2 = fmaf(vx.y, vx.y, sx2);
      sx2 = fmaf(vx.z, vx.z, sx2); sx2 = fmaf(vx.w, vx.w, sx2);
      sy2 = fmaf(vy.x, vy.x, sy2); sy2 = fmaf(vy.y, vy.y, sy2);
      sy2 = fmaf(vy.z, vy.z, sy2); sy2 = fmaf(vy.w, vy.w, sy2);
      sxy = fmaf(vx.x, vy.x, sxy); sxy = fmaf(vx.y, vy.y, sxy);
      sxy = fmaf(vx.z, vy.z, sxy); sxy = fmaf(vx.w, vy.w, sxy);
    }
    sx2 = wave_allsum(sx2);
    sy2 = wave_allsum(sy2);
    sxy = wave_allsum(sxy);
    if (lane == 0) {
      gram_sh[wave][0] = sx2;
      gram_sh[wave][1] = sy2;
      gram_sh[wave][2] = sxy;
    }
  }
  __syncthreads();

  // ---------- scalar hyperbolic chain: 8 rows lane-parallel on wave 0 ----
  if (tid < WPB && ((int)blockIdx.x * WPB + tid) < rows) {
    const Gram g{gram_sh[tid][0], gram_sh[tid][1], gram_sh[tid][2]};

    const float c     = log1pf(expf(curv_hra[0]));                 // softplus
    const float sc    = sqrtf(c);
    const float s     = 1.0f / (1.0f + expf(-scale_center[0]));    // sigmoid
    const float gmax  = 1.0f + log1pf(expf(gamma_scale[0]));
    const float gamma = gmax * tanhf(gamma_raw[0]);
    const float maxn_pre  = 15.0f / sc;
    const float maxn_post = (1.0f - 1e-5f) / sc;

    const V2 X{1.0f, 0.0f}, Y{0.0f, 1.0f};
    V2 hrx = clipn(expmap0_c(clipn(X, g, maxn_pre), g, sc), g, maxn_post);
    V2 hry = clipn(expmap0_c(clipn(Y, g, maxn_pre), g, sc), g, maxn_post);
    V2 sx  = clipn(msm(s, hrx, g, sc), g, maxn_post);
    V2 sy  = clipn(msm(1.0f - s, hry, g, sc), g, maxn_post);
    V2 p   = clipn(mobius_add_c(sx, sy, g, c), g, maxn_post);
    V2 np{-p.a, -p.b};
    V2 xp  = clipn(mobius_add_c(np, hrx, g, c), g, maxn_post);
    V2 yp  = clipn(mobius_add_c(np, hry, g, c), g, maxn_post);
    V2 ys  = clipn(msm(gamma, yp, g, sc), g, maxn_post);
    V2 hp  = clipn(mobius_add_c(xp, ys, g, c), g, maxn_post);
    V2 hr  = mobius_add_c(p, hp, g, c);
    V2 r   = clipn(hr, g, maxn_post);
    V2 o   = logmap0_c(r, g, sc);

    coef_sh[tid][0] = o.a;
    coef_sh[tid][1] = o.b;
  }
  __syncthreads();

  // ---------- out = alpha*x + beta*y, streamed from LDS ------------------
  if (live) {
    const float A = coef_sh[wave][0];
    const float B = coef_sh[wave][1];
    float* orow = out + (size_t)row * D;
#pragma unroll
    for (int j = 0; j < CHUNKS; ++j) {
      const int e = j * 128 + lane * 4;
      float4 vx = *(const float4*)&sh[xoff + e];
      float4 vy = *(const float4*)&sh[yoff + e];
      float4 o;
      o.x = fmaf(A, vx.x, B * vy.x);
      o.y = fmaf(A, vx.y, B * vy.y);
      o.z = fmaf(A, vx.z, B * vy.z);
      o.w = fmaf(A, vx.w, B * vy.w);
      *(float4*)(orow + e) = o;
    }
  }
}

} // anonymous namespace

extern "C" void kernel_launch(void* const* d_in, const int* in_sizes, int n_in,
                              void* d_out, int out_size, void* d_ws, size_t ws_size,
                              hipStream_t stream) {
  const float* x       = (const float*)d_in[0];
  const float* y       = (const float*)d_in[1];
  const float* curv    = (const float*)d_in[2];
  const float* graw    = (const float*)d_in[3];
  const float* gscale  = (const float*)d_in[4];
  const float* scenter = (const float*)d_in[5];
  float* out = (float*)d_out;

  const int rows   = in_sizes[0] / D;                 // 64*577 = 36928
  const int blocks = (rows + WPB - 1) / WPB;          // 4616, exact
  hra_kernel<<<dim3(blocks), dim3(WPB * LANES), 0, stream>>>(
      x, y, curv, graw, gscale, scenter, out, rows);
}